// multi_adaptive_hypergraoh_13984413516354
// MI455X (gfx1250) — compile-verified
//
#include <hip/hip_runtime.h>
#include <hip/hip_bf16.h>
#include <stdint.h>

typedef __attribute__((ext_vector_type(2))) float v2f;
typedef __attribute__((ext_vector_type(4))) float v4f;
typedef __attribute__((ext_vector_type(8))) float v8f;

#define D_MODEL 1024
#define H_NUM   1024   // hyperedges per layer
#define TILE_M  16     // node rows per workgroup
#define WAVES   8      // waves per workgroup (wave32)
#define CTILES_PER_WAVE 8   // 64 column tiles of 16 / 8 waves

// One workgroup handles TILE_M node rows x all 1024 hyperedges.
// Wave w owns column tiles [8w, 8w+8) (columns [128w, 128w+128)).
__global__ __launch_bounds__(256) void hyperlayer_kernel(const float* __restrict__ node,
                                                         const float* __restrict__ hyp,
                                                         float* __restrict__ out) {
    // A-tile in LDS, pre-swizzled to WMMA fragment order:
    // float2 index = kq*32 + lane, holding node[(row0 + lane%16)*D + 4*kq + 2*(lane/16)]
    __shared__ v2f sA[TILE_M * D_MODEL / 2];           // 64 KB
    __shared__ float sM[WAVES][TILE_M];
    __shared__ float sZ[WAVES][TILE_M];
    __shared__ int   sArg[WAVES][TILE_M];
    __shared__ int   sFinalArg[TILE_M];
    __shared__ float sFinalEmit[TILE_M];

    const int tid  = threadIdx.x;
    const int wave = tid >> 5;
    const int lane = tid & 31;
    const int llo  = lane & 15;       // lane % 16
    const int lhi  = lane >> 4;       // 0 or 1
    const long row0 = (long)blockIdx.x * TILE_M;

    // ---- A-tile staging: per-lane swizzled gather, global -> LDS via the
    //      CDNA5 async-copy path (no VGPR round trip, tracked by ASYNCcnt) ----
    {
        const int nchunks = TILE_M * D_MODEL / 2;      // 8192 float2 chunks
        #pragma unroll
        for (int it = 0; it < nchunks / 256; ++it) {
            int chunk = tid + it * 256;
            int kq  = chunk >> 5;
            int ln  = chunk & 31;
            int row = ln & 15;
            int col = (kq << 2) + ((ln >> 4) << 1);
            const float* gaddr = node + (row0 + row) * D_MODEL + col;
            // flat LDS pointer: addr[31:0] is the LDS byte offset (aperture rule)
            uint32_t ldsOff = (uint32_t)(uintptr_t)(&sA[chunk]);
            asm volatile("global_load_async_to_lds_b64 %0, %1, off"
                         :
                         : "v"(ldsOff), "v"(gaddr)
                         : "memory");
        }
        asm volatile("s_wait_asynccnt 0x0" ::: "memory");
    }
    __syncthreads();

    // ---- GEMM: 8 accumulators of 16x16, K=1024 in steps of 4 via fp32 WMMA ----
    v8f c[CTILES_PER_WAVE];
    #pragma unroll
    for (int j = 0; j < CTILES_PER_WAVE; ++j) c[j] = (v8f)(0.0f);

    // B fragment base: hyp row (col0 + lane%16), k-offset 2*(lane/16)
    const float* hypBase = hyp + (long)(wave * 128 + llo) * D_MODEL + (lhi << 1);

    #pragma unroll 4
    for (int kq = 0; kq < D_MODEL / 4; ++kq) {
        v2f a = sA[(kq << 5) + lane];                  // ds_load_b64, conflict-free
        const float* hp = hypBase + (kq << 2);
        #pragma unroll
        for (int j = 0; j < CTILES_PER_WAVE; ++j) {
            v2f b = *(const v2f*)(hp + j * 16 * D_MODEL);
            c[j] = __builtin_amdgcn_wmma_f32_16x16x4_f32(
                       false, a, false, b, (short)0, c[j], false, false);
        }
    }

    // ---- per-row max/argmax/sum-exp over this wave's 128 columns ----
    // C layout: component v, lanes 0-15 -> row v, lanes 16-31 -> row v+8; col = 16j + lane%16
    #pragma unroll
    for (int v = 0; v < 8; ++v) {
        float bm = -1.0f; int bi = 0;
        #pragma unroll
        for (int j = 0; j < CTILES_PER_WAVE; ++j) {
            float s = fmaxf(0.0f, 3.0f * c[j][v]);
            if (s > bm) { bm = s; bi = wave * 128 + j * 16 + llo; }
        }
        // 16-lane (half-wave) max+arg reduction
        #pragma unroll
        for (int off = 1; off < 16; off <<= 1) {
            float om = __shfl_xor(bm, off, 32);
            int   oi = __shfl_xor(bi, off, 32);
            if (om > bm) { bm = om; bi = oi; }
        }
        // sum of exp(s - rowmax)
        float z = 0.0f;
        #pragma unroll
        for (int j = 0; j < CTILES_PER_WAVE; ++j) {
            float s = fmaxf(0.0f, 3.0f * c[j][v]);
            z += __expf(s - bm);
        }
        #pragma unroll
        for (int off = 1; off < 16; off <<= 1) z += __shfl_xor(z, off, 32);

        if (llo == 0) {
            int row = v + (lhi << 3);
            sM[wave][row]   = bm;
            sZ[wave][row]   = z;
            sArg[wave][row] = bi;
        }
    }
    __syncthreads();

    // ---- cross-wave combine (softmax max/Z merge); emit iff 1/Z > 0.5 ----
    if (tid < TILE_M) {
        int r = tid;
        float M = -1.0f; int A = 0;
        #pragma unroll
        for (int w = 0; w < WAVES; ++w) {
            float mw = sM[w][r];
            if (mw > M) { M = mw; A = sArg[w][r]; }
        }
        float Zt = 0.0f;
        #pragma unroll
        for (int w = 0; w < WAVES; ++w) Zt += sZ[w][r] * __expf(sM[w][r] - M);
        sFinalArg[r]  = A;
        sFinalEmit[r] = (Zt < 2.0f) ? 1.0f : 0.0f;   // max softmax = 1/Z > 0.5
    }
    __syncthreads();

    // ---- dense output write: 16 rows x 1024 cols, coalesced nontemporal float4 ----
    const int colb = tid << 2;                         // 256 threads x 4 = 1024
    #pragma unroll
    for (int r = 0; r < TILE_M; ++r) {
        int   A = sFinalArg[r];
        float e = sFinalEmit[r];
        v4f v;
        v.x = (A == colb + 0) ? e : 0.0f;
        v.y = (A == colb + 1) ? e : 0.0f;
        v.z = (A == colb + 2) ? e : 0.0f;
        v.w = (A == colb + 3) ? e : 0.0f;
        __builtin_nontemporal_store(v, (v4f*)(out + (row0 + r) * D_MODEL + colb));
    }
}

extern "C" void kernel_launch(void* const* d_in, const int* in_sizes, int n_in,
                              void* d_out, int out_size, void* d_ws, size_t ws_size,
                              hipStream_t stream) {
    // inputs (setup_inputs order): x, hyp_emb_0..2, node_emb_0..2
    const float* hyp0  = (const float*)d_in[1];
    const float* hyp1  = (const float*)d_in[2];
    const float* hyp2  = (const float*)d_in[3];
    const float* node0 = (const float*)d_in[4];
    const float* node1 = (const float*)d_in[5];
    const float* node2 = (const float*)d_in[6];

    float* out0 = (float*)d_out;
    float* out1 = out0 + (long)32768 * 1024;
    float* out2 = out1 + (long)8192 * 1024;

    hyperlayer_kernel<<<32768 / TILE_M, 256, 0, stream>>>(node0, hyp0, out0);
    hyperlayer_kernel<<< 8192 / TILE_M, 256, 0, stream>>>(node1, hyp1, out1);
    hyperlayer_kernel<<< 2048 / TILE_M, 256, 0, stream>>>(node2, hyp2, out2);
}